// PolyLoss_25838523253233
// MI455X (gfx1250) — compile-verified
//
#include <hip/hip_runtime.h>
#include <hip/hip_bf16.h>

typedef __attribute__((ext_vector_type(2))) float v2f;
typedef __attribute__((ext_vector_type(8))) float v8f;

#define PV    16          // vertices per polygon
#define PMAXN 64          // MAXN = 4*V
#define PB    16
#define PK    128
#define PHW   (256 * 256)
#define PC    (2 * PV)

__device__ __forceinline__ float safef(float d) { return (d == 0.0f) ? 1.0f : d; }

// Reproduces reference _area_padded exactly, including the double-counted
// v0 -> v1 edge term.
__device__ float area_padded(const float* x, const float* y, int n) {
    float left = 0.0f, right = 0.0f;
    for (int i = 0; i < n; ++i) {
        int j = (i + 1) % n;              // n >= 1 when loop body runs
        left  += x[i] * y[j];
        right += y[i] * x[j];
    }
    if (n > 0) {
        int i1 = 1 % n;
        left  += x[0] * y[i1];
        right += y[0] * x[i1];
    }
    return fabsf(0.5f * (right - left));
}

// One thread per (b,k) polygon pair: gather pred poly, Sutherland-Hodgman clip
// against target poly, areas, IoU. Writes iou*mask to workspace.
__global__ __launch_bounds__(256)
void poly_iou_kernel(const float* __restrict__ output,
                     const float* __restrict__ mask,
                     const int*   __restrict__ ind,
                     const float* __restrict__ target,
                     float* __restrict__ ws_iou) {
    int tid = blockIdx.x * blockDim.x + threadIdx.x;
    if (tid >= PB * PK) return;
    int b = tid / PK;

    // ---- gather predicted polygon: pred[v] = output[b, 2v(+1), ind] ----
    float sx[PV], sy[PV];                 // subject (pred) polygon
    float tx[PV], ty[PV];                 // clip (target) polygon
    {
        const float* obase = output + (size_t)b * PC * PHW + (size_t)ind[tid];
        const float* tbase = target + (size_t)tid * (2 * PV);
        #pragma unroll
        for (int v = 0; v < PV; ++v) {
            sx[v] = obase[(size_t)(2 * v) * PHW];
            sy[v] = obase[(size_t)(2 * v + 1) * PHW];
            tx[v] = tbase[2 * v];
            ty[v] = tbase[2 * v + 1];
        }
    }

    // ---- Sutherland-Hodgman clip (16 edges, poly capped at PMAXN) ----
    float cpx[PMAXN], cpy[PMAXN];         // current polygon
    float nqx[PMAXN], nqy[PMAXN];         // next polygon
    int n = PV;
    #pragma unroll 1
    for (int v = 0; v < PV; ++v) { cpx[v] = sx[v]; cpy[v] = sy[v]; }

    #pragma unroll 1
    for (int e = 0; e < PV; ++e) {
        float c1x = tx[(e + PV - 1) & (PV - 1)], c1y = ty[(e + PV - 1) & (PV - 1)];
        float c2x = tx[e],                       c2y = ty[e];
        float ex = c2x - c1x, ey = c2y - c1y;
        int m = 0;
        #pragma unroll 1
        for (int i = 0; i < n; ++i) {
            int pi = (i == 0) ? (n - 1) : (i - 1);
            float pxx = cpx[pi], pyy = cpy[pi];
            float cxx = cpx[i],  cyy = cpy[i];
            bool ins_c = (ex * (cyy - c1y) - ey * (cxx - c1x)) <= 0.0f;
            bool ins_p = (ex * (pyy - c1y) - ey * (pxx - c1x)) <= 0.0f;
            if (ins_c != ins_p) {
                // _intersection(prev, cur, c1, c2), reference semantics
                float dx12 = cxx - pxx, dy12 = cyy - pyy;
                float dx34 = ex;
                float m1 = dy12 / safef(dx12);
                float b1 = pyy - m1 * pxx;
                float m2 = ey / safef(dx34);
                float b2 = c1y - m2 * c1x;
                float xi, yi;
                if (dx12 == 0.0f) {
                    xi = pxx;             yi = m2 * pxx + b2;
                } else if (dx34 == 0.0f) {
                    xi = c1x;             yi = m1 * c1x + b1;
                } else {
                    xi = (b2 - b1) / safef(m1 - m2);
                    yi = m1 * xi + b1;
                }
                if (m < PMAXN) { nqx[m] = xi; nqy[m] = yi; }
                ++m;
            }
            if (ins_c) {
                if (m < PMAXN) { nqx[m] = cxx; nqy[m] = cyy; }
                ++m;
            }
        }
        n = (m < PMAXN) ? m : PMAXN;
        #pragma unroll 1
        for (int i = 0; i < n; ++i) { cpx[i] = nqx[i]; cpy[i] = nqy[i]; }
    }

    // ---- areas + IoU ----
    float a_i = area_padded(cpx, cpy, n);
    float a_p = area_padded(sx, sy, PV);
    float a_g = area_padded(tx, ty, PV);
    float inter = ((a_i == 0.0f) ? fminf(a_p, a_g) : 0.0f) + a_i;
    float uni   = a_g + a_p - inter;
    float iou   = inter / (uni + 1e-6f);

    ws_iou[tid] = iou * mask[tid];
}

// Single wave32 reduction of 2048 values (iou*mask) and 2048 mask values using
// V_WMMA_F32_16X16X4_F32 with an all-ones A matrix: D = 1*B + C accumulates
// column sums of each 64-value chunk; every row of D is identical, so the full
// 32-lane sum of C's VGPR0 equals 2 * total.
__global__ __launch_bounds__(32)
void reduce_wmma_kernel(const float* __restrict__ vals,
                        const float* __restrict__ mask,
                        float* __restrict__ out, int n) {
    int lane = threadIdx.x;               // 0..31, EXEC all ones (WMMA requirement)
    v2f a; a[0] = 1.0f; a[1] = 1.0f;      // A = ones(16x4)
    v8f ci = {};                          // Σ iou*mask accumulator
    v8f cm = {};                          // Σ mask accumulator
    #pragma unroll 1
    for (int i = 0; i < n; i += 64) {
        v2f bi, bm;                       // B = 4x16 chunk (2 VGPRs x 32 lanes)
        bi[0] = vals[i + lane];
        bi[1] = vals[i + 32 + lane];
        bm[0] = mask[i + lane];
        bm[1] = mask[i + 32 + lane];
        ci = __builtin_amdgcn_wmma_f32_16x16x4_f32(false, a, false, bi,
                                                   (short)0, ci, false, false);
        cm = __builtin_amdgcn_wmma_f32_16x16x4_f32(false, a, false, bm,
                                                   (short)0, cm, false, false);
    }
    float si = ci[0];                     // C VGPR0: colsum(lane%16), duplicated twice
    float sm = cm[0];
    #pragma unroll
    for (int off = 16; off >= 1; off >>= 1) {
        si += __shfl_xor(si, off, 32);
        sm += __shfl_xor(sm, off, 32);
    }
    si *= 0.5f;                           // each column sum counted twice across 32 lanes
    sm *= 0.5f;
    if (lane == 0)
        out[0] = 1.0f - si / (sm + 1e-6f);
}

extern "C" void kernel_launch(void* const* d_in, const int* in_sizes, int n_in,
                              void* d_out, int out_size, void* d_ws, size_t ws_size,
                              hipStream_t stream) {
    const float* output = (const float*)d_in[0];  // (16, 32, 256, 256) f32
    const float* mask   = (const float*)d_in[1];  // (16, 128) f32
    const int*   ind    = (const int*)  d_in[2];  // (16, 128) i32
    const float* target = (const float*)d_in[3];  // (16, 128, 32) f32
    float* out = (float*)d_out;                   // scalar loss
    float* ws_iou = (float*)d_ws;                 // 2048 floats of scratch

    const int NPOLY = PB * PK;                    // 2048
    poly_iou_kernel<<<(NPOLY + 255) / 256, 256, 0, stream>>>(
        output, mask, ind, target, ws_iou);
    reduce_wmma_kernel<<<1, 32, 0, stream>>>(ws_iou, mask, out, NPOLY);
}